// SepsisClassifier_79104707657773
// MI455X (gfx1250) — compile-verified
//
#include <hip/hip_runtime.h>
#include <math.h>

// ACE-NODE Euler scan, persistent-kernel formulation for gfx1250 (MI455X).
// B=256, T=256, H=512. Three [256x512]x[512x512] fp32 GEMMs per step via
// V_WMMA_F32_16X16X4_F32. y/x tiles staged in LDS with GLOBAL_LOAD_ASYNC_TO_LDS
// (ASYNCcnt), x(t+1) prefetched across the grid barrier while step t computes.

#define Bn   256
#define Tn   256
#define Hn   512
#define DTs  0.1f
#define NBLK 32
#define NTHR 256   // 8 waves of 32

typedef __attribute__((ext_vector_type(2))) float v2f;
typedef __attribute__((ext_vector_type(8))) float v8f;

__global__ void ace_bar_init(unsigned* __restrict__ bar) {
  bar[0] = 0u;  // arrival counter
  bar[1] = 0u;  // generation
}

__device__ __forceinline__ void grid_barrier(unsigned* cnt, unsigned* gen) {
  __syncthreads();
  if (threadIdx.x == 0) {
    unsigned g = __atomic_load_n(gen, __ATOMIC_RELAXED);
    __threadfence();  // make this block's global stores visible device-wide
    unsigned prev = __atomic_fetch_add(cnt, 1u, __ATOMIC_ACQ_REL);
    if (prev == (unsigned)(NBLK - 1)) {
      __atomic_store_n(cnt, 0u, __ATOMIC_RELAXED);
      __atomic_fetch_add(gen, 1u, __ATOMIC_ACQ_REL);  // release new generation
    } else {
      while (__atomic_load_n(gen, __ATOMIC_ACQUIRE) == g) {
        __builtin_amdgcn_s_sleep(8);
      }
    }
  }
  __syncthreads();
}

// Async-copy a 32x512 fp32 tile (64KB) global -> LDS using
// GLOBAL_LOAD_ASYNC_TO_LDS_B128. 4096 16B chunks over 256 threads (16 each).
// Completion is tracked by ASYNCcnt (s_wait_asynccnt).
__device__ __forceinline__ void async_copy_tile(const float* gbase,
                                                size_t row_stride_elts,
                                                uint32_t lds_base_off,
                                                int tidb) {
  #pragma unroll
  for (int j = 0; j < 16; ++j) {
    const int c = tidb + (j << 8);          // chunk id 0..4095
    const int row = c >> 7;                 // 128 chunks per 2KB row
    const int colc = c & 127;
    const uint64_t ga = (uint64_t)(uintptr_t)(gbase + (size_t)row * row_stride_elts)
                        + ((uint64_t)colc << 4);
    const uint32_t la = lds_base_off + ((uint32_t)c << 4);
    asm volatile("global_load_async_to_lds_b128 %0, %1, off"
                 :: "v"(la), "v"(ga) : "memory");
  }
}

__device__ __forceinline__ void wait_async0() {
  asm volatile("s_wait_asynccnt 0x0" ::: "memory");
}

__global__ void __launch_bounds__(NTHR)
ace_node_kernel(const float* __restrict__ x,    // [B,T,H]
                const float* __restrict__ y0,   // [B,H]
                const float* __restrict__ W,    // [H,H]
                const float* __restrict__ U,    // [H,H]
                const float* __restrict__ b,    // [H]
                const float* __restrict__ Aw,   // [H*H] -> [H,H] row-major
                const float* __restrict__ Wro,  // [1,H]
                const float* __restrict__ bro,  // [1]
                float* __restrict__ out,        // [B,T]
                float* __restrict__ ybuf0,      // [B,H] ws
                float* __restrict__ ybuf1,      // [B,H] ws
                unsigned* __restrict__ bar)     // 2 u32 ws
{
  __shared__ __align__(16) float ytile[32 * Hn];      // 64KB  : y rows m0..m0+31
  __shared__ __align__(16) float xtile[2][32 * Hn];   // 128KB : x(t) double buffer

  const int tidb = threadIdx.x;
  const int tid  = blockIdx.x * NTHR + tidb;

  // ---- tile assignment: 256 waves, each owns M=32 x N=16 of y ----
  const int lane = threadIdx.x & 31;
  const int wv   = (blockIdx.x << 3) + (threadIdx.x >> 5);  // 0..255
  const int mg   = wv >> 5;   // 0..7  -> m0 = 32*mg  (same for all waves in block)
  const int nt   = wv & 31;   // 0..31 -> n0 = 16*nt
  const int m0   = mg * 32;
  const int n0   = nt * 16;
  const int l15  = lane & 15;
  const int khalf = (lane >> 4) << 1;     // lanes 16-31 hold K+2,K+3 (ISA 16x4 f32 layout)
  const int nrow  = n0 + l15;             // weight row (output column)

  const uint32_t lds_y  = (uint32_t)(uintptr_t)&ytile[0];
  const uint32_t lds_x0 = (uint32_t)(uintptr_t)&xtile[0][0];
  const uint32_t lds_x1 = (uint32_t)(uintptr_t)&xtile[1][0];

  // ---- prefetch x(0) tile immediately (overlaps with init phase) ----
  async_copy_tile(x + (size_t)m0 * Tn * Hn, (size_t)Tn * Hn, lds_x0, tidb);

  // ---- phase 0: init state buffer + output (harness poisons buffers) ----
  const float bias_ro = bro[0];
  for (int i = tid; i < Bn * Hn; i += NBLK * NTHR) ybuf0[i] = y0[i];
  for (int i = tid; i < Bn * Tn; i += NBLK * NTHR) out[i] = bias_ro;
  grid_barrier(bar, bar + 1);

  const float bn    = b[nrow];
  const float wro_n = Wro[nrow];
  const float* Wrow = W  + (size_t)nrow * Hn;
  const float* Urow = U  + (size_t)nrow * Hn;
  const float* Arow = Aw + (size_t)nrow * Hn;
  const int mlo = (lane >> 4) << 3;       // C/D layout: lanes>=16 hold M+8
  // LDS fragment rows (local row index within the 32-row tile)
  const float* yl0 = ytile + (size_t)l15 * Hn;
  const float* yl1 = ytile + (size_t)(16 + l15) * Hn;

  for (int t = 0; t < Tn; ++t) {
    float* ycur = (t & 1) ? ybuf1 : ybuf0;
    float* ynxt = (t & 1) ? ybuf0 : ybuf1;

    // stage y(t) tile (just published by the previous grid barrier)
    async_copy_tile(ycur + (size_t)m0 * Hn, (size_t)Hn, lds_y, tidb);
    wait_async0();        // completes y(t) stage AND x(t) prefetch (per-wave ASYNCcnt)
    __syncthreads();      // LDS tiles visible to all waves in the block

    // prefetch x(t+1) into the other buffer; stays in flight through this step
    if (t + 1 < Tn) {
      async_copy_tile(x + (size_t)m0 * Tn * Hn + (size_t)(t + 1) * Hn,
                      (size_t)Tn * Hn, (t & 1) ? lds_x0 : lds_x1, tidb);
    }
    const float* xl = xtile[t & 1];
    const float* xl0 = xl + (size_t)l15 * Hn;
    const float* xl1 = xl + (size_t)(16 + l15) * Hn;

    v8f accp0, accp1, accg0, accg1;
    #pragma unroll
    for (int r = 0; r < 8; ++r) { accp0[r] = bn; accp1[r] = bn; accg0[r] = 0.f; accg1[r] = 0.f; }

    #pragma unroll 4
    for (int k = 0; k < Hn; k += 4) {
      const int kk = k + khalf;
      v2f bwf = *(const v2f*)(Wrow + kk);   // global (L2-resident weights)
      v2f buf = *(const v2f*)(Urow + kk);
      v2f baf = *(const v2f*)(Arow + kk);
      v2f ay0 = *(const v2f*)(yl0 + kk);    // LDS (ds_load_b64)
      v2f ay1 = *(const v2f*)(yl1 + kk);
      v2f ax0 = *(const v2f*)(xl0 + kk);
      v2f ax1 = *(const v2f*)(xl1 + kk);
      accp0 = __builtin_amdgcn_wmma_f32_16x16x4_f32(false, ay0, false, bwf, (short)0, accp0, false, false);
      accg0 = __builtin_amdgcn_wmma_f32_16x16x4_f32(false, ay0, false, baf, (short)0, accg0, false, false);
      accp0 = __builtin_amdgcn_wmma_f32_16x16x4_f32(false, ax0, false, buf, (short)0, accp0, false, false);
      accp1 = __builtin_amdgcn_wmma_f32_16x16x4_f32(false, ay1, false, bwf, (short)0, accp1, false, false);
      accg1 = __builtin_amdgcn_wmma_f32_16x16x4_f32(false, ay1, false, baf, (short)0, accg1, false, false);
      accp1 = __builtin_amdgcn_wmma_f32_16x16x4_f32(false, ax1, false, buf, (short)0, accp1, false, false);
    }

    // ---- elementwise update + fused readout ----
    #pragma unroll
    for (int s = 0; s < 2; ++s) {
      v8f& accp = s ? accp1 : accp0;
      v8f& accg = s ? accg1 : accg0;
      const int mloc_base = 16 * s + mlo;   // local row in the 32-row tile
      #pragma unroll
      for (int r = 0; r < 8; ++r) {
        const int mloc = mloc_base + r;
        const int m = m0 + mloc;
        const float gate = 1.0f / (1.0f + __expf(-accg[r]));
        const float yold = ytile[(size_t)mloc * Hn + nrow];   // LDS read
        const float yn   = yold + DTs * tanhf(accp[r]) * gate;
        ynxt[(size_t)m * Hn + nrow] = yn;
        // partial readout: sum over the 16 N-columns this wave owns for row m
        float val = yn * wro_n;
        val += __shfl_xor(val, 1, 32);
        val += __shfl_xor(val, 2, 32);
        val += __shfl_xor(val, 4, 32);
        val += __shfl_xor(val, 8, 32);
        if (l15 == 0) atomicAdd(&out[(size_t)m * Tn + t], val);
      }
    }

    grid_barrier(bar, bar + 1);  // publish y_{t+1}; also fences ytile reuse
  }
}

extern "C" void kernel_launch(void* const* d_in, const int* in_sizes, int n_in,
                              void* d_out, int out_size, void* d_ws, size_t ws_size,
                              hipStream_t stream) {
  const float* x   = (const float*)d_in[0];
  const float* y0  = (const float*)d_in[1];
  const float* W   = (const float*)d_in[2];
  const float* U   = (const float*)d_in[3];
  const float* b   = (const float*)d_in[4];
  const float* A   = (const float*)d_in[5];
  const float* Wro = (const float*)d_in[6];
  const float* bro = (const float*)d_in[7];
  float* out = (float*)d_out;

  float* ybuf0 = (float*)d_ws;
  float* ybuf1 = ybuf0 + (size_t)Bn * Hn;
  unsigned* bar = (unsigned*)(ybuf1 + (size_t)Bn * Hn);

  ace_bar_init<<<1, 1, 0, stream>>>(bar);
  ace_node_kernel<<<NBLK, NTHR, 0, stream>>>(x, y0, W, U, b, A, Wro, bro,
                                             out, ybuf0, ybuf1, bar);
}